// GraphChannelMixerPyG_11184094838957
// MI455X (gfx1250) — compile-verified
//
#include <hip/hip_runtime.h>

typedef __attribute__((ext_vector_type(2))) float v2f;
typedef __attribute__((ext_vector_type(8))) float v8f;

#define ALPHA_C 0.05f
#define CCO     0.475f      /* (1-ALPHA)/K_HOPS */
#define NN      19
#define NP      32
#define DD      64
#define F0      80
#define TT      1024
#define LAP_EPS 1e-4f
#define NSWEEP  7

// ---- WMMA f32 16x16x4 fragment helpers (ISA 7.12.2 layouts) ----
// A (16x4, MxK): lanes 0-15 -> row M=l, VGPR0=K0,VGPR1=K1; lanes 16-31 -> K2,K3
__device__ __forceinline__ v2f lda_frag(const float* A, int lda, int m0, int k, int lane) {
  int l = lane & 15;
  int h = (lane >> 4) << 1;
  v2f a;
  a.x = A[(m0 + l) * lda + k + h];
  a.y = A[(m0 + l) * lda + k + h + 1];
  return a;
}
// B (4x16, KxN): lanes 0-15 -> col N=l, rows K0/K1; lanes 16-31 -> rows K2/K3
__device__ __forceinline__ v2f ldb_frag(const float* B, int ldb, int k, int n0, int lane) {
  int l = lane & 15;
  int h = (lane >> 4) << 1;
  v2f b;
  b.x = B[(k + h) * ldb + n0 + l];
  b.y = B[(k + h + 1) * ldb + n0 + l];
  return b;
}

__device__ __forceinline__ v8f wmma_step(v2f a, v2f b, v8f c) {
  return __builtin_amdgcn_wmma_f32_16x16x4_f32(false, a, false, b, (short)0, c,
                                               false, false);
}

// ---- CDNA5 async memory <-> LDS (ASYNCcnt-tracked DMA, bypasses VGPRs) ----
__device__ __forceinline__ void async_ld_b128(unsigned lds, const void* g) {
  asm volatile("global_load_async_to_lds_b128 %0, %1, off"
               :: "v"(lds), "v"(g) : "memory");
}
__device__ __forceinline__ void async_ld_b32(unsigned lds, const void* g) {
  asm volatile("global_load_async_to_lds_b32 %0, %1, off"
               :: "v"(lds), "v"(g) : "memory");
}
__device__ __forceinline__ void async_st_b128(void* g, unsigned lds) {
  asm volatile("global_store_async_from_lds_b128 %0, %1, off"
               :: "v"(g), "v"(lds) : "memory");
}
__device__ __forceinline__ void wait_async0() {
  asm volatile("s_wait_asynccnt 0x0" ::: "memory");
}

__global__ __launch_bounds__(256)
void gcm_kernel(const float* __restrict__ feat, const float* __restrict__ adjm,
                const float* __restrict__ edge_w, const float* __restrict__ edge_b,
                const float* __restrict__ W0, const float* __restrict__ b0,
                const float* __restrict__ W1, const float* __restrict__ b1,
                const float* __restrict__ g0, const float* __restrict__ be0,
                const float* __restrict__ g1, const float* __restrict__ be1,
                float* __restrict__ out)
{
  __shared__ __align__(16) float sA[NN * NN];   // raw adjacency (async DMA target)
  __shared__ float sWp[NN * NN];                // softplus-weighted W (+I)
  __shared__ float sQ[NP * 20];                 // Q = P^T, padded 32x20
  __shared__ float sL[NN * 20];                 // Jacobi workspace
  __shared__ float sV[NN * 20];                 // eigenvector accumulator
  __shared__ __align__(16) float sX0[NP * 84];  // x0 = [x | pe], stride 84 (336B, 16B-aligned rows)
  __shared__ __align__(16) float sT1[NP * 84];
  __shared__ __align__(16) float sT2[NP * 84];  // final result (async DMA source)
  __shared__ float sH[NP * 68];                 // linear0 output / LN0, stride 68
  __shared__ float sDisW[NN];
  __shared__ float sDisL[NN];
  __shared__ float sEv[NN];
  __shared__ float sSgn[16];
  __shared__ int   sIdx[NN];
  __shared__ float sc[10], ss[10];

  const int tid  = threadIdx.x;
  const int lane = tid & 31;
  const int wave = tid >> 5;
  const int g  = blockIdx.x;
  const int bI = g >> 10;           // TT = 1024
  const int tI = g & (TT - 1);

  const unsigned ldsA  = (unsigned)(uintptr_t)sA;    // low 32b of flat addr = LDS offset
  const unsigned ldsX0 = (unsigned)(uintptr_t)sX0;
  const unsigned ldsT2 = (unsigned)(uintptr_t)sT2;

  if (tid == 0) {                    // weights are hot across all 16384 blocks
    __builtin_prefetch(W0, 0, 3);
    __builtin_prefetch(W1, 0, 3);
  }

  // ---- kick off adjacency DMA (needed first) ----
  const size_t abase = (size_t)g * (NN * NN);
  for (int i = tid; i < NN * NN; i += 256)
    async_ld_b32(ldsA + 4u * (unsigned)i, &adjm[abase + i]);

  // ---- zero only the padding of x0 (disjoint from the feature-DMA region) ----
  for (int i = tid; i < NP * 20; i += 256) {      // cols 64..83, all 32 rows
    int r = i / 20, c = i % 20;
    sX0[r * 84 + 64 + c] = 0.f;
  }
  for (int i = tid; i < 13 * DD; i += 256) {      // rows 19..31, cols 0..63
    int r = i >> 6, c = i & 63;
    sX0[(NN + r) * 84 + c] = 0.f;
  }
  for (int i = tid; i < NP * 20; i += 256) sQ[i] = 0.f;

  wait_async0();                     // adjacency resident in LDS
  __syncthreads();

  // ---- kick off feature DMA; it drains only after the Jacobi solver ----
  for (int i = tid; i < NN * 16; i += 256) {      // 19 rows x 16 chunks of 16B
    int n = i >> 4, c = i & 15;
    size_t off = (((size_t)(bI * NN + n)) * TT + tI) * DD + c * 4;
    async_ld_b128(ldsX0 + (unsigned)(n * 336 + c * 16), &feat[off]);
  }

  // ---- propagation weights: W = softplus(a*ew+eb) masked, +I ----
  const float ew = edge_w[0], eb = edge_b[0];
  for (int i = tid; i < NN * NN; i += 256) {
    int r = i / NN, c = i % NN;
    float a = sA[i];
    float z = a * ew + eb;
    float w = (z > 20.f) ? z : log1pf(expf(z));
    w = (a > 0.f) ? w : 0.f;
    if (r == c) w += 1.f;
    sWp[i] = w;
  }
  __syncthreads();

  // ---- degrees: deg_W = column sums of W; deg_L = row sums of a ----
  if (tid < NN) {
    int j = tid;
    float dw = 0.f, dl = 0.f;
    for (int i = 0; i < NN; ++i) { dw += sWp[i * NN + j]; dl += sA[j * NN + i]; }
    sDisW[j] = (dw > 0.f) ? rsqrtf(dw) : 0.f;
    sDisL[j] = rsqrtf(fmaxf(dl, LAP_EPS));
  }
  __syncthreads();

  // ---- Q = P^T (Q[j][i] = dis[i]*W[i][j]*dis[j]); symmetrized Laplacian; V=I ----
  for (int i = tid; i < NN * NN; i += 256) {
    int r = i / NN, c = i % NN;
    sQ[c * 20 + r] = sDisW[r] * sWp[i] * sDisW[c];
    float off = sDisL[r] * sDisL[c] * 0.5f * (sA[r * NN + c] + sA[c * NN + r]);
    sL[r * 20 + c] = ((r == c) ? (1.f + LAP_EPS) : 0.f) - off;
    sV[r * 20 + c] = (r == c) ? 1.f : 0.f;
  }
  __syncthreads();

  // ---- parallel-pairs cyclic Jacobi: 19 rounds x 9 disjoint rotations ----
  for (int sweep = 0; sweep < NSWEEP; ++sweep) {
    for (int rr = 0; rr < NN; ++rr) {
      if (tid < 9) {
        int k = tid + 1;
        int p = (rr + k) % NN, q = (rr + NN - k) % NN;
        float apq = sL[p * 20 + q];
        float cc = 1.f, sv = 0.f;
        if (fabsf(apq) > 1e-12f) {
          float theta = (sL[q * 20 + q] - sL[p * 20 + p]) / (2.f * apq);
          float tt = copysignf(1.f, theta) / (fabsf(theta) + sqrtf(theta * theta + 1.f));
          cc = 1.f / sqrtf(tt * tt + 1.f);
          sv = tt * cc;
        }
        sc[tid] = cc; ss[tid] = sv;
      }
      __syncthreads();
      if (tid < 9 * NN) {            // row phase: L <- J^T L
        int pr = tid / NN, col = tid % NN;
        int k = pr + 1;
        int p = (rr + k) % NN, q = (rr + NN - k) % NN;
        float cc = sc[pr], sv = ss[pr];
        float lp = sL[p * 20 + col], lq = sL[q * 20 + col];
        sL[p * 20 + col] = cc * lp - sv * lq;
        sL[q * 20 + col] = sv * lp + cc * lq;
      }
      __syncthreads();
      if (tid < 9 * NN) {            // col phase: L <- L J ; V <- V J
        int pr = tid / NN, row = tid % NN;
        int k = pr + 1;
        int p = (rr + k) % NN, q = (rr + NN - k) % NN;
        float cc = sc[pr], sv = ss[pr];
        float lp = sL[row * 20 + p], lq = sL[row * 20 + q];
        sL[row * 20 + p] = cc * lp - sv * lq;
        sL[row * 20 + q] = sv * lp + cc * lq;
        float vp = sV[row * 20 + p], vq = sV[row * 20 + q];
        sV[row * 20 + p] = cc * vp - sv * vq;
        sV[row * 20 + q] = sv * vp + cc * vq;
      }
      __syncthreads();
    }
  }

  // ---- sort eigenvalues ascending, sign-fix, write PE into x0[:,64:80] ----
  if (tid < NN) sEv[tid] = sL[tid * 20 + tid];
  __syncthreads();
  if (tid == 0) {
    for (int i = 0; i < NN; ++i) sIdx[i] = i;
    for (int a = 0; a < NN - 1; ++a) {
      int best = a;
      for (int b = a + 1; b < NN; ++b)
        if (sEv[sIdx[b]] < sEv[sIdx[best]]) best = b;
      int tmp = sIdx[a]; sIdx[a] = sIdx[best]; sIdx[best] = tmp;
    }
  }
  __syncthreads();
  if (tid < 16) {
    int ci = sIdx[tid];
    float sum = 0.f;
    for (int i = 0; i < NN; ++i) sum += sV[i * 20 + ci];
    sSgn[tid] = (sum > 0.f) ? 1.f : ((sum < 0.f) ? -1.f : 1.f);
  }
  __syncthreads();
  for (int i = tid; i < NN * 16; i += 256) {
    int n = i >> 4, m = i & 15;
    float v = sV[n * 20 + sIdx[m]] * sSgn[m];
    if (v != v) v = 0.f;
    else if (v > 3.0e38f) v = 1.f;
    else if (v < -3.0e38f) v = -1.f;
    sX0[n * 84 + 64 + m] = v;
  }

  wait_async0();                     // features (DMA'd during Jacobi) now resident
  __syncthreads();

  // ==================== conv block 0 (F=80) ====================
  // t1 = Q @ x0   (M=32/2 tiles, N=80/5 tiles, K=20)
  for (int tile = wave; tile < 10; tile += 8) {
    int m0 = (tile / 5) * 16, n0 = (tile % 5) * 16;
    v8f acc = {0.f, 0.f, 0.f, 0.f, 0.f, 0.f, 0.f, 0.f};
    for (int k = 0; k < 20; k += 4)
      acc = wmma_step(lda_frag(sQ, 20, m0, k, lane), ldb_frag(sX0, 84, k, n0, lane), acc);
    int l = lane & 15, h8 = (lane >> 4) * 8;
    for (int r = 0; r < 8; ++r) sT1[(m0 + r + h8) * 84 + n0 + l] = acc[r];
  }
  __syncthreads();
  // t2 = Q @ t1
  for (int tile = wave; tile < 10; tile += 8) {
    int m0 = (tile / 5) * 16, n0 = (tile % 5) * 16;
    v8f acc = {0.f, 0.f, 0.f, 0.f, 0.f, 0.f, 0.f, 0.f};
    for (int k = 0; k < 20; k += 4)
      acc = wmma_step(lda_frag(sQ, 20, m0, k, lane), ldb_frag(sT1, 84, k, n0, lane), acc);
    int l = lane & 15, h8 = (lane >> 4) * 8;
    for (int r = 0; r < 8; ++r) sT2[(m0 + r + h8) * 84 + n0 + l] = acc[r];
  }
  __syncthreads();
  // h0 = alpha*x0 + 0.475*(t1+t2)  (in place in sX0)
  for (int i = tid; i < NP * F0; i += 256) {
    int r = i / F0, c = i % F0;
    int o = r * 84 + c;
    sX0[o] = ALPHA_C * sX0[o] + CCO * (sT1[o] + sT2[o]);
  }
  __syncthreads();
  // y = h0 @ W0 + b0 -> sH   (K=80; W0 read straight from global, hits WGP$/L2)
  for (int tile = wave; tile < 8; tile += 8) {
    int m0 = (tile / 4) * 16, n0 = (tile % 4) * 16;
    v8f acc = {0.f, 0.f, 0.f, 0.f, 0.f, 0.f, 0.f, 0.f};
    for (int k = 0; k < F0; k += 4)
      acc = wmma_step(lda_frag(sX0, 84, m0, k, lane), ldb_frag(W0, 64, k, n0, lane), acc);
    int l = lane & 15, h8 = (lane >> 4) * 8;
    float bias = b0[n0 + l];
    for (int r = 0; r < 8; ++r) sH[(m0 + r + h8) * 68 + n0 + l] = acc[r] + bias;
  }
  __syncthreads();
  // LayerNorm 0 (rows < 19)
  if (tid < NN) {
    float* row = &sH[tid * 68];
    float mu = 0.f;
    for (int c = 0; c < DD; ++c) mu += row[c];
    mu *= (1.f / DD);
    float var = 0.f;
    for (int c = 0; c < DD; ++c) { float d = row[c] - mu; var += d * d; }
    var *= (1.f / DD);
    float inv = rsqrtf(var + 1e-5f);
    for (int c = 0; c < DD; ++c) row[c] = (row[c] - mu) * inv * g0[c] + be0[c];
  }
  __syncthreads();

  // ==================== conv block 1 (F=64) + residual ====================
  // t1 = Q @ xb
  for (int tile = wave; tile < 8; tile += 8) {
    int m0 = (tile / 4) * 16, n0 = (tile % 4) * 16;
    v8f acc = {0.f, 0.f, 0.f, 0.f, 0.f, 0.f, 0.f, 0.f};
    for (int k = 0; k < 20; k += 4)
      acc = wmma_step(lda_frag(sQ, 20, m0, k, lane), ldb_frag(sH, 68, k, n0, lane), acc);
    int l = lane & 15, h8 = (lane >> 4) * 8;
    for (int r = 0; r < 8; ++r) sT1[(m0 + r + h8) * 84 + n0 + l] = acc[r];
  }
  __syncthreads();
  // t2 = Q @ t1
  for (int tile = wave; tile < 8; tile += 8) {
    int m0 = (tile / 4) * 16, n0 = (tile % 4) * 16;
    v8f acc = {0.f, 0.f, 0.f, 0.f, 0.f, 0.f, 0.f, 0.f};
    for (int k = 0; k < 20; k += 4)
      acc = wmma_step(lda_frag(sQ, 20, m0, k, lane), ldb_frag(sT1, 84, k, n0, lane), acc);
    int l = lane & 15, h8 = (lane >> 4) * 8;
    for (int r = 0; r < 8; ++r) sT2[(m0 + r + h8) * 84 + n0 + l] = acc[r];
  }
  __syncthreads();
  // h1 = alpha*xb + 0.475*(t1+t2) -> sT1
  for (int i = tid; i < NP * DD; i += 256) {
    int r = i >> 6, c = i & 63;
    sT1[r * 84 + c] = ALPHA_C * sH[r * 68 + c] + CCO * (sT1[r * 84 + c] + sT2[r * 84 + c]);
  }
  __syncthreads();
  // y2 = h1 @ W1 + b1 + xb (residual) -> sT2
  for (int tile = wave; tile < 8; tile += 8) {
    int m0 = (tile / 4) * 16, n0 = (tile % 4) * 16;
    v8f acc = {0.f, 0.f, 0.f, 0.f, 0.f, 0.f, 0.f, 0.f};
    for (int k = 0; k < DD; k += 4)
      acc = wmma_step(lda_frag(sT1, 84, m0, k, lane), ldb_frag(W1, 64, k, n0, lane), acc);
    int l = lane & 15, h8 = (lane >> 4) * 8;
    float bias = b1[n0 + l];
    for (int r = 0; r < 8; ++r) {
      int row = m0 + r + h8;
      sT2[row * 84 + n0 + l] = acc[r] + bias + sH[row * 68 + n0 + l];
    }
  }
  __syncthreads();
  // LayerNorm 1 (rows < 19) in sT2
  if (tid < NN) {
    float* row = &sT2[tid * 84];
    float mu = 0.f;
    for (int c = 0; c < DD; ++c) mu += row[c];
    mu *= (1.f / DD);
    float var = 0.f;
    for (int c = 0; c < DD; ++c) { float d = row[c] - mu; var += d * d; }
    var *= (1.f / DD);
    float inv = rsqrtf(var + 1e-5f);
    for (int c = 0; c < DD; ++c) row[c] = (row[c] - mu) * inv * g1[c] + be1[c];
  }
  __syncthreads();

  // ---- write output (B,N,T,D) straight from LDS via async DMA ----
  for (int i = tid; i < NN * 16; i += 256) {      // 19 rows x 16 chunks of 16B
    int n = i >> 4, c = i & 15;
    size_t off = (((size_t)(bI * NN + n)) * TT + tI) * DD + c * 4;
    async_st_b128((void*)&out[off], ldsT2 + (unsigned)(n * 336 + c * 16));
  }
  wait_async0();   // (S_ENDPGM also drains, but be explicit)
}

extern "C" void kernel_launch(void* const* d_in, const int* in_sizes, int n_in,
                              void* d_out, int out_size, void* d_ws, size_t ws_size,
                              hipStream_t stream) {
  (void)in_sizes; (void)n_in; (void)out_size; (void)d_ws; (void)ws_size;
  const float* feat = (const float*)d_in[0];
  const float* adjm = (const float*)d_in[1];
  const float* ew   = (const float*)d_in[2];
  const float* eb   = (const float*)d_in[3];
  const float* W0   = (const float*)d_in[4];
  const float* b0   = (const float*)d_in[5];
  const float* W1   = (const float*)d_in[6];
  const float* b1   = (const float*)d_in[7];
  const float* g0   = (const float*)d_in[8];
  const float* be0  = (const float*)d_in[9];
  const float* g1   = (const float*)d_in[10];
  const float* be1  = (const float*)d_in[11];

  dim3 grid(16 * 1024);   // G = B*T = 16384 graphs, one workgroup each
  dim3 block(256);        // 8 waves (wave32)
  gcm_kernel<<<grid, block, 0, stream>>>(feat, adjm, ew, eb, W0, b0, W1, b1,
                                         g0, be0, g1, be1, (float*)d_out);
}